// HGTConv_47888885351096
// MI455X (gfx1250) — compile-verified
//
#include <hip/hip_runtime.h>
#include <hip/hip_bf16.h>

typedef float v2f __attribute__((ext_vector_type(2)));
typedef float v8f __attribute__((ext_vector_type(8)));

#define HEADS 8
#define DDIM 16
#define CH 128   // IN == OUT == H*D == 128

// ---------------------------------------------------------------------------
// Fold per-head relation matrix into a 128x128 weight (and its bias):
//   Wf[i, h*16+e] = sum_d W[i, h*16+d] * rel[h, d, e]
//   bf[h*16+e]    = sum_d b[h*16+d]    * rel[h, d, e]
// ---------------------------------------------------------------------------
__global__ void fold_rel_kernel(const float* __restrict__ W,
                                const float* __restrict__ b,
                                const float* __restrict__ rel,
                                float* __restrict__ Wf,
                                float* __restrict__ bf) {
  int j = blockIdx.x * blockDim.x + threadIdx.x;   // 0 .. 128*128-1
  if (j >= CH * CH) return;
  int col = j & (CH - 1);
  int i   = j >> 7;
  int h   = col >> 4;
  int e2  = col & 15;
  const float* relh = rel + h * (DDIM * DDIM) + e2;   // rel[h][d][e2], stride 16 over d
  const float* wrow = W + (size_t)i * CH + h * DDIM;
  float acc = 0.f;
#pragma unroll
  for (int d = 0; d < DDIM; ++d) acc += wrow[d] * relh[d * DDIM];
  Wf[j] = acc;
  if (i == 0) {
    const float* brow = b + h * DDIM;
    float accb = 0.f;
#pragma unroll
    for (int d = 0; d < DDIM; ++d) accb += brow[d] * relh[d * DDIM];
    bf[col] = accb;
  }
}

// ---------------------------------------------------------------------------
// C[M,128] = A[M,128] x B[128,128] + bias, optional residual blend epilogue:
//   if skip:  out = sigmoid(skip)*(acc+bias) + (1-sigmoid(skip))*resid
// One wave per 16x16 C tile via V_WMMA_F32_16X16X4_F32; 8 waves/block cover a
// 16-row strip (N = 128 = 8 tiles). M must be a multiple of 16 (50000 = 3125*16).
// ---------------------------------------------------------------------------
__global__ __launch_bounds__(256) void gemm128_wmma_kernel(
    const float* __restrict__ A, const float* __restrict__ B,
    const float* __restrict__ bias, float* __restrict__ C,
    const float* __restrict__ skip, const float* __restrict__ resid, int M) {
  const int lane   = threadIdx.x & 31;
  const int wv     = threadIdx.x >> 5;          // 0..7 -> N tile
  const int tile_m = blockIdx.x * 16;
  const int tile_n = wv * 16;
  const int half   = lane >> 4;                 // 0|1 selects K sub-pair
  const int l15    = lane & 15;

  v8f acc = {};
  const float* arow = A + (size_t)(tile_m + l15) * CH;
#pragma unroll
  for (int k0 = 0; k0 < CH; k0 += 4) {
    const int kk = k0 + half * 2;
    v2f af, bf_;
    af.x  = arow[kk];
    af.y  = arow[kk + 1];
    bf_.x = B[(size_t)kk * CH + tile_n + l15];
    bf_.y = B[(size_t)(kk + 1) * CH + tile_n + l15];
    acc = __builtin_amdgcn_wmma_f32_16x16x4_f32(
        /*neg_a=*/false, af, /*neg_b=*/false, bf_,
        /*c_mod=*/(short)0, acc, /*reuse_a=*/false, /*reuse_b=*/false);
  }

  const int   col = tile_n + l15;
  const float bv  = bias[col];
  float al = 1.f, bl = 0.f;
  if (skip != nullptr) {                         // uniform branch, EXEC unchanged
    const float s = 1.f / (1.f + expf(-skip[0]));
    al = s; bl = 1.f - s;
  }
#pragma unroll
  for (int r = 0; r < 8; ++r) {
    const int row = tile_m + r + half * 8;
    float val = acc[r] + bv;
    if (resid != nullptr) val = al * val + bl * resid[(size_t)row * CH + col];
    C[(size_t)row * CH + col] = val;
  }
}

// float -> order-preserving uint key (total order for non-NaN)
__device__ __forceinline__ unsigned orderKey(float f) {
  unsigned u = __float_as_uint(f);
  return (u & 0x80000000u) ? ~u : (u | 0x80000000u);
}
__device__ __forceinline__ float decodeKey(unsigned k) {
  unsigned u = (k & 0x80000000u) ? (k ^ 0x80000000u) : ~k;
  return __uint_as_float(u);
}

// ---------------------------------------------------------------------------
// Pass 1: one wave32 per edge. lane -> head = lane>>2, 4 floats per lane.
// logits[e,h] = dot(q[tgt,h,:], k[src,h,:]) * prel[h] * D^-0.5 ; atomicMax.
// ---------------------------------------------------------------------------
__global__ void edge_logits_max_kernel(const float* __restrict__ q,
                                       const float* __restrict__ k,
                                       const int* __restrict__ src,
                                       const int* __restrict__ tgt,
                                       const float* __restrict__ prel,
                                       float* __restrict__ elog,
                                       unsigned* __restrict__ maxkey, int E) {
  const int wave = (int)((blockIdx.x * (size_t)blockDim.x + threadIdx.x) >> 5);
  const int lane = threadIdx.x & 31;
  if (wave >= E) return;
  const int s = src[wave], t = tgt[wave];
  const int h = lane >> 2, sub = lane & 3;
  const size_t off = (size_t)h * DDIM + sub * 4;
  const float4 q4 = *(const float4*)(q + (size_t)t * CH + off);
  const float4 k4 = *(const float4*)(k + (size_t)s * CH + off);
  float dot = q4.x * k4.x + q4.y * k4.y + q4.z * k4.z + q4.w * k4.w;
  dot += __shfl_xor(dot, 1);
  dot += __shfl_xor(dot, 2);                       // all 4 lanes of a head hold the sum
  if (sub == 0) {
    const float lg = dot * prel[h] * 0.25f;        // 1/sqrt(16)
    elog[(size_t)wave * HEADS + h] = lg;
    atomicMax(&maxkey[(size_t)t * HEADS + h], orderKey(lg));
  }
}

// ---------------------------------------------------------------------------
// Pass 2: one thread per (edge, head): e = exp(logit - max); accumulate denom.
// ---------------------------------------------------------------------------
__global__ void edge_exp_den_kernel(float* __restrict__ elog,
                                    const int* __restrict__ tgt,
                                    const unsigned* __restrict__ maxkey,
                                    float* __restrict__ den, int EH) {
  const int i = blockIdx.x * blockDim.x + threadIdx.x;
  if (i >= EH) return;
  const int e = i >> 3, h = i & 7;
  const int t = tgt[e];
  const float m  = decodeKey(maxkey[(size_t)t * HEADS + h]);
  const float ex = expf(elog[i] - m);
  elog[i] = ex;
  atomicAdd(&den[(size_t)t * HEADS + h], ex);
}

// ---------------------------------------------------------------------------
// Pass 3: one wave32 per edge; att = e/(den+1e-16); scatter-add att * v[src].
// ---------------------------------------------------------------------------
__global__ void edge_aggregate_kernel(const float* __restrict__ v,
                                      const float* __restrict__ elog,
                                      const float* __restrict__ den,
                                      const int* __restrict__ src,
                                      const int* __restrict__ tgt,
                                      float* __restrict__ agg, int E) {
  const int wave = (int)((blockIdx.x * (size_t)blockDim.x + threadIdx.x) >> 5);
  const int lane = threadIdx.x & 31;
  if (wave >= E) return;
  const int s = src[wave], t = tgt[wave];
  const int h = lane >> 2, sub = lane & 3;
  const float att = elog[(size_t)wave * HEADS + h] /
                    (den[(size_t)t * HEADS + h] + 1e-16f);
  const size_t off = (size_t)h * DDIM + sub * 4;
  const float4 v4 = *(const float4*)(v + (size_t)s * CH + off);
  float* dst = agg + (size_t)t * CH + off;
  atomicAdd(dst + 0, v4.x * att);
  atomicAdd(dst + 1, v4.y * att);
  atomicAdd(dst + 2, v4.z * att);
  atomicAdd(dst + 3, v4.w * att);
}

// ---------------------------------------------------------------------------
extern "C" void kernel_launch(void* const* d_in, const int* in_sizes, int n_in,
                              void* d_out, int out_size, void* d_ws, size_t ws_size,
                              hipStream_t stream) {
  const float* x_a   = (const float*)d_in[0];
  const float* x_b   = (const float*)d_in[1];
  const float* qW_a  = (const float*)d_in[2];
  const float* qb_a  = (const float*)d_in[3];
  const float* kW_a  = (const float*)d_in[4];
  const float* kb_a  = (const float*)d_in[5];
  const float* vW_a  = (const float*)d_in[6];
  const float* vb_a  = (const float*)d_in[7];
  const float* aW_a  = (const float*)d_in[8];
  const float* ab_a  = (const float*)d_in[9];
  const float* skp_a = (const float*)d_in[10];
  const float* qW_b  = (const float*)d_in[11];
  const float* qb_b  = (const float*)d_in[12];
  const float* kW_b  = (const float*)d_in[13];
  const float* kb_b  = (const float*)d_in[14];
  const float* vW_b  = (const float*)d_in[15];
  const float* vb_b  = (const float*)d_in[16];
  const float* aW_b  = (const float*)d_in[17];
  const float* ab_b  = (const float*)d_in[18];
  const float* skp_b = (const float*)d_in[19];
  const float* arel_ab = (const float*)d_in[20];
  const float* mrel_ab = (const float*)d_in[21];
  const float* prel_ab = (const float*)d_in[22];
  const float* arel_ba = (const float*)d_in[23];
  const float* mrel_ba = (const float*)d_in[24];
  const float* prel_ba = (const float*)d_in[25];
  const int* src_ab = (const int*)d_in[26];
  const int* tgt_ab = (const int*)d_in[27];
  const int* src_ba = (const int*)d_in[28];
  const int* tgt_ba = (const int*)d_in[29];

  const int NA  = in_sizes[0] / CH;
  const int NB  = in_sizes[1] / CH;
  const int Eab = in_sizes[26];
  const int Eba = in_sizes[28];

  // ---- workspace carve-out (floats) ----
  float* ws = (float*)d_ws;
  size_t o = 0;
  auto carve = [&](size_t n) { float* p = ws + o; o += n; return p; };
  float* q_a   = carve((size_t)NA * CH);
  float* q_b   = carve((size_t)NB * CH);
  float* k_ab  = carve((size_t)NA * CH);
  float* v_ab  = carve((size_t)NA * CH);
  float* k_ba  = carve((size_t)NB * CH);
  float* v_ba  = carve((size_t)NB * CH);
  float* agg_a = carve((size_t)NA * CH);
  float* agg_b = carve((size_t)NB * CH);
  float* e_ab  = carve((size_t)Eab * HEADS);
  float* e_ba  = carve((size_t)Eba * HEADS);
  unsigned* max_b = (unsigned*)carve((size_t)NB * HEADS);
  unsigned* max_a = (unsigned*)carve((size_t)NA * HEADS);
  float* den_b = carve((size_t)NB * HEADS);
  float* den_a = carve((size_t)NA * HEADS);
  float* Wk_ab = carve(CH * CH); float* bk_ab = carve(CH);
  float* Wv_ab = carve(CH * CH); float* bv_ab = carve(CH);
  float* Wk_ba = carve(CH * CH); float* bk_ba = carve(CH);
  float* Wv_ba = carve(CH * CH); float* bv_ba = carve(CH);

  // ---- fold relation matrices into K/V weights ----
  const int foldBlocks = (CH * CH + 255) / 256;
  fold_rel_kernel<<<foldBlocks, 256, 0, stream>>>(kW_a, kb_a, arel_ab, Wk_ab, bk_ab);
  fold_rel_kernel<<<foldBlocks, 256, 0, stream>>>(vW_a, vb_a, mrel_ab, Wv_ab, bv_ab);
  fold_rel_kernel<<<foldBlocks, 256, 0, stream>>>(kW_b, kb_b, arel_ba, Wk_ba, bk_ba);
  fold_rel_kernel<<<foldBlocks, 256, 0, stream>>>(vW_b, vb_b, mrel_ba, Wv_ba, bv_ba);

  // ---- projection GEMMs (WMMA f32) ----
  const int gA = NA / 16, gB = NB / 16;
  gemm128_wmma_kernel<<<gA, 256, 0, stream>>>(x_a, qW_a, qb_a, q_a, nullptr, nullptr, NA);
  gemm128_wmma_kernel<<<gB, 256, 0, stream>>>(x_b, qW_b, qb_b, q_b, nullptr, nullptr, NB);
  gemm128_wmma_kernel<<<gA, 256, 0, stream>>>(x_a, Wk_ab, bk_ab, k_ab, nullptr, nullptr, NA);
  gemm128_wmma_kernel<<<gA, 256, 0, stream>>>(x_a, Wv_ab, bv_ab, v_ab, nullptr, nullptr, NA);
  gemm128_wmma_kernel<<<gB, 256, 0, stream>>>(x_b, Wk_ba, bk_ba, k_ba, nullptr, nullptr, NB);
  gemm128_wmma_kernel<<<gB, 256, 0, stream>>>(x_b, Wv_ba, bv_ba, v_ba, nullptr, nullptr, NB);

  // ---- zero accumulators (graph-capturable async memsets) ----
  hipMemsetAsync(agg_a, 0, (size_t)NA * CH * sizeof(float), stream);
  hipMemsetAsync(agg_b, 0, (size_t)NB * CH * sizeof(float), stream);
  hipMemsetAsync(den_a, 0, (size_t)NA * HEADS * sizeof(float), stream);
  hipMemsetAsync(den_b, 0, (size_t)NB * HEADS * sizeof(float), stream);
  hipMemsetAsync(max_a, 0, (size_t)NA * HEADS * sizeof(unsigned), stream);  // key 0 == -inf-ish
  hipMemsetAsync(max_b, 0, (size_t)NB * HEADS * sizeof(unsigned), stream);

  // ---- edge phase: direction a->b aggregates into b (q from b, k/v from a) ----
  const int ebABblk = (int)(((size_t)Eab * 32 + 255) / 256);
  const int ebBAblk = (int)(((size_t)Eba * 32 + 255) / 256);
  edge_logits_max_kernel<<<ebABblk, 256, 0, stream>>>(q_b, k_ab, src_ab, tgt_ab,
                                                      prel_ab, e_ab, max_b, Eab);
  edge_logits_max_kernel<<<ebBAblk, 256, 0, stream>>>(q_a, k_ba, src_ba, tgt_ba,
                                                      prel_ba, e_ba, max_a, Eba);

  const int exABblk = (Eab * HEADS + 255) / 256;
  const int exBAblk = (Eba * HEADS + 255) / 256;
  edge_exp_den_kernel<<<exABblk, 256, 0, stream>>>(e_ab, tgt_ab, max_b, den_b, Eab * HEADS);
  edge_exp_den_kernel<<<exBAblk, 256, 0, stream>>>(e_ba, tgt_ba, max_a, den_a, Eba * HEADS);

  edge_aggregate_kernel<<<ebABblk, 256, 0, stream>>>(v_ab, e_ab, den_b, src_ab, tgt_ab,
                                                     agg_b, Eab);
  edge_aggregate_kernel<<<ebBAblk, 256, 0, stream>>>(v_ba, e_ba, den_a, src_ba, tgt_ba,
                                                     agg_a, Eba);

  // ---- output GEMMs with sigmoid-skip residual epilogue ----
  float* out_a = (float*)d_out;
  float* out_b = (float*)d_out + (size_t)NA * CH;
  gemm128_wmma_kernel<<<gA, 256, 0, stream>>>(agg_a, aW_a, ab_a, out_a, skp_a, x_a, NA);
  gemm128_wmma_kernel<<<gB, 256, 0, stream>>>(agg_b, aW_b, ab_b, out_b, skp_b, x_b, NB);
}